// Shift_15298673508386
// MI455X (gfx1250) — compile-verified
//
#include <hip/hip_runtime.h>

// Shift op: out[n,c,y,x] = in[n,c,y+dy_c,x+dx_c], zero padded. Pure bandwidth.
// B=32, C=256, H=W=56, fp32. ~205 MB total traffic -> ~8.8us floor @ 23.3 TB/s.

#define BB 32
#define CC 256
#define HH 56
#define WW 56
#define PLANE (HH * WW)          // 3136 floats = 12544 bytes (16B multiple)
#define NV4   (PLANE / 4)        // 784 float4 per plane

#if defined(__gfx1250__) && __has_builtin(__builtin_amdgcn_global_load_async_to_lds_b128)
#define USE_ASYNC_LDS 1
#else
#define USE_ASYNC_LDS 0
#endif

// b128 async builtin wants int4-typed, address-space-qualified pointers:
//   arg0: AS(1) ("__device__") int4*  (global source)
//   arg1: AS(3) ("__shared__") int4*  (LDS destination)
typedef int v4i __attribute__((ext_vector_type(4)));
typedef __attribute__((address_space(1))) v4i GV4;   // global int4
typedef __attribute__((address_space(3))) v4i LV4;   // LDS int4

__global__ __launch_bounds__(256) void shift_kernel(const float* __restrict__ in,
                                                    const int*   __restrict__ shift,
                                                    float*       __restrict__ out) {
    const int plane = blockIdx.x;            // n*C + c
    const int c     = plane % CC;
    // Per-channel shift: workgroup-uniform -> SGPRs via s_load.
    const int dy = shift[2 * c + 0];
    const int dx = shift[2 * c + 1];

    const float* __restrict__ ip = in  + (size_t)plane * PLANE;
    float*       __restrict__ op = out + (size_t)plane * PLANE;

    __shared__ float tile[PLANE];

    // ---- Stage full plane global -> LDS ------------------------------------
#if USE_ASYNC_LDS
    for (int v = threadIdx.x; v < NV4; v += blockDim.x) {
        __builtin_amdgcn_global_load_async_to_lds_b128(
            (GV4*)(ip + 4 * v),
            (LV4*)(tile + 4 * v),
            /*offset=*/0, /*cpol=*/0);
    }
#if __has_builtin(__builtin_amdgcn_s_wait_asynccnt)
    __builtin_amdgcn_s_wait_asynccnt(0);
#else
    asm volatile("s_wait_asynccnt 0" ::: "memory");
#endif
    __syncthreads();
#else
    for (int v = threadIdx.x; v < NV4; v += blockDim.x) {
        ((float4*)tile)[v] = ((const float4*)ip)[v];
    }
    __syncthreads();
#endif

    // ---- Shifted write-out: aligned float4 stores, LDS gathers -------------
    for (int v = threadIdx.x; v < NV4; v += blockDim.x) {
        const int lin = 4 * v;
        const int y   = lin / WW;
        const int x0  = lin % WW;           // W=56 divisible by 4 -> rows never split
        const int sy  = y + dy;

        float4 r = make_float4(0.f, 0.f, 0.f, 0.f);
        if ((unsigned)sy < (unsigned)HH) {
            const float* row = tile + sy * WW;
            float e[4];
#pragma unroll
            for (int k = 0; k < 4; ++k) {
                int sx  = x0 + k + dx;
                int cx  = sx < 0 ? 0 : (sx > WW - 1 ? WW - 1 : sx);  // clamp: safe LDS addr
                float t = row[cx];                                    // ds_load_b32
                e[k] = ((unsigned)sx < (unsigned)WW) ? t : 0.f;       // v_cndmask edge mask
            }
            r = make_float4(e[0], e[1], e[2], e[3]);
        }
        ((float4*)op)[v] = r;               // global_store_b128
    }
}

extern "C" void kernel_launch(void* const* d_in, const int* in_sizes, int n_in,
                              void* d_out, int out_size, void* d_ws, size_t ws_size,
                              hipStream_t stream) {
    const float* x     = (const float*)d_in[0];   // [B,C,H,W] fp32
    const int*   shift = (const int*)d_in[1];     // [C,2] int32
    float*       out   = (float*)d_out;

    dim3 grid(BB * CC);   // one block per (n,c) plane
    dim3 block(256);      // 8 wave32
    shift_kernel<<<grid, block, 0, stream>>>(x, shift, out);
}